// DSINE_v02_kappa_74354473828548
// MI455X (gfx1250) — compile-verified
//
#include <hip/hip_runtime.h>
#include <hip/hip_bf16.h>
#include <math.h>

// ---------------------------------------------------------------------------
// DSINE forward, fp32, conv-as-implicit-GEMM via V_WMMA_F32_16X16X4_F32.
// A (im2col) staged in LDS cooperatively; B (weights) staged in LDS via
// gfx1250 async global->LDS loads (ASYNCcnt). 4 waves / block.
// B=4, H=60, W=80, HID=FEAT=64, PS=5, PAD=2, K=8, NUM_ITER=2.
// ---------------------------------------------------------------------------

typedef __attribute__((ext_vector_type(2))) float v2f;
typedef __attribute__((ext_vector_type(8))) float v8f;

#define BB   4
#define HH   60
#define WW   80
#define HWC  4800            // H*W
#define MTOT 19200           // B*H*W
#define MT   1200            // MTOT/16
#define UPH  480
#define UPW  640
#define PRED_STRIDE (BB*4*UPH*UPW)   // 4,915,200

#define KC   64              // K-chunk staged in LDS per iteration
#define LPAD 4               // row pad (floats): 272B rows, bank-stride 4

// activation codes
#define ACT_NONE 0
#define ACT_RELU 1
#define ACT_SIG  2
#define ACT_TANH 3

__device__ __forceinline__ float act_apply(float v, int act) {
  if (act == ACT_RELU) return fmaxf(v, 0.0f);
  if (act == ACT_SIG)  return 1.0f / (1.0f + expf(-v));
  if (act == ACT_TANH) return tanhf(v);
  return v;
}

// ---------------------------------------------------------------------------
// Implicit-GEMM conv: out(B,Cout,H,W) = act(conv(in, wgt(Cout,Cin,KS,KS)) + bias)
//
// Block = 128 threads = 4 waves. blockIdx.x = M tile (16 spatial positions),
// blockIdx.y selects a 64-wide slab of output channels; wave w owns columns
// blockIdx.y*64 + w*16 .. +15.
//
// Per K-chunk (KC=64):
//   - As[16][KC] im2col slice staged with normal stores (zero-padded past K)
//   - Bs[64][KC] weight slice staged with global_load_async_to_lds_b64
//     (weights contiguous in K per output channel; 8B alignment guaranteed)
//   - s_wait_asynccnt 0 (own ops) then barrier -> all-wave visibility
//   - 16 x v_wmma_f32_16x16x4_f32, A/B fragments via conflict-free ds_load_b64
//
// Fragment layouts (ISA 7.12.2, 32-bit, wave32):
//   A 16x4 : lanes 0-15 rows x K{0,1}; lanes 16-31 rows x K{2,3}
//   B 4x16 : lanes 0-15 cols x K{0,1}; lanes 16-31 cols x K{2,3}
//   C/D    : VGPR j -> (M=j, N=lane) / (M=j+8, N=lane-16)
// ---------------------------------------------------------------------------
template<int KS, int PADC>
__global__ __launch_bounds__(128)
void conv_wmma(const float* __restrict__ in, const float* __restrict__ wgt,
               const float* __restrict__ bias, float* __restrict__ out,
               int Cin, int Cout, int act)
{
  __shared__ float As[16][KC + LPAD];
  __shared__ float Bs[64][KC + LPAD];

  const int tid  = threadIdx.x;
  const int lane = tid & 31;
  const int wave = tid >> 5;         // 0..3
  const int l16  = lane & 15;
  const int half = lane >> 4;        // 0 -> K{0,1}, 1 -> K{2,3}
  const int mt   = blockIdx.x;
  const int Ktot = Cin * KS * KS;

  const int  ncol   = blockIdx.y * 64 + wave * 16 + l16;
  const int  lcol   = wave * 16 + l16;          // column inside Bs
  const bool nvalid = (ncol < Cout);
  const long wbase  = (long)ncol * (long)Ktot;

  v8f acc = {0.f, 0.f, 0.f, 0.f, 0.f, 0.f, 0.f, 0.f};

  for (int k0 = 0; k0 < Ktot; k0 += KC) {
    __syncthreads();               // protect As/Bs reads from previous chunk
    // ---- cooperative im2col staging: 16 rows x KC cols, zero-padded ----
#pragma unroll
    for (int e = tid; e < 16 * KC; e += 128) {
      const int r  = e >> 6;       // KC == 64
      const int kk = e & (KC - 1);
      const int k  = k0 + kk;
      float a = 0.0f;
      if (k < Ktot) {
        const int m   = mt * 16 + r;
        const int b   = m / HWC;
        const int pos = m - b * HWC;
        const int y   = pos / WW;
        const int x   = pos - y * WW;
        const int ci  = k / (KS * KS);
        const int rem = k - ci * (KS * KS);
        const int ky  = rem / KS;
        const int kx  = rem - ky * KS;
        const int yy  = y + ky - PADC;
        const int xx  = x + kx - PADC;
        if (yy >= 0 && yy < HH && xx >= 0 && xx < WW)
          a = in[((b * Cin + ci) * HH + yy) * WW + xx];
      }
      As[r][kk] = a;
    }

    const bool full = (k0 + KC) <= Ktot;
    if (full) {
      // ---- async weight staging: 64 cols x KC floats, 8B per op ----
      // e indexes 8-byte units: 64 cols * 32 units = 2048 ops / 128 thr = 16 ea
#pragma unroll
      for (int e = tid; e < 64 * (KC / 2); e += 128) {
        const int col  = e >> 5;           // KC/2 == 32
        const int seg  = e & 31;
        const int colg = blockIdx.y * 64 + col;
        if (colg < Cout) {
          const float* src = &wgt[(long)colg * (long)Ktot + k0 + seg * 2];
          unsigned dst = (unsigned)(size_t)&Bs[col][seg * 2];
          asm volatile("global_load_async_to_lds_b64 %0, %1, off"
                       :: "v"(dst), "v"(src) : "memory");
        }
      }
      // wait for this wave's async LDS writes, then make them block-visible
      asm volatile("s_wait_asynccnt 0x0" ::: "memory");
      // prefetch next weight chunk toward L2/WGP (global_prefetch_b8)
      if (nvalid && (k0 + KC) < Ktot)
        __builtin_prefetch(&wgt[wbase + k0 + KC + half * 32], 0, 0);
    }
    __syncthreads();

#pragma unroll 4
    for (int kk = 0; kk < KC; kk += 4) {
      v2f av, bv;
      // A fragment from LDS: two consecutive floats -> ds_load_b64
      const float2 a2 = *(const float2*)&As[l16][kk + half * 2];
      av[0] = a2.x; av[1] = a2.y;
      if (full) {
        const float2 w2 = *(const float2*)&Bs[lcol][kk + half * 2];
        bv[0] = nvalid ? w2.x : 0.f;
        bv[1] = nvalid ? w2.y : 0.f;
      } else {
        const int k = k0 + kk + half * 2;
        bv[0] = (nvalid && k     < Ktot) ? wgt[wbase + k]     : 0.f;
        bv[1] = (nvalid && k + 1 < Ktot) ? wgt[wbase + k + 1] : 0.f;
      }
      acc = __builtin_amdgcn_wmma_f32_16x16x4_f32(false, av, false, bv,
                                                  (short)0, acc, false, false);
    }
  }

  if (!nvalid) return;
  const float bval = bias[ncol];
#pragma unroll
  for (int j = 0; j < 8; ++j) {
    const int mr = mt * 16 + half * 8 + j;
    const int b2 = mr / HWC;
    const int p2 = mr - b2 * HWC;
    float v = acc[j] + bval;
    v = act_apply(v, act);
    out[(b2 * Cout + ncol) * HWC + p2] = v;
  }
}

// ---------------------------------------------------------------------------
// Elementwise kernels
// ---------------------------------------------------------------------------

// uv8(B,2,H,W): camera-normalized pixel coords
__global__ void k_uv8(const float* __restrict__ intr, float* __restrict__ uv8) {
  int i = blockIdx.x * blockDim.x + threadIdx.x;
  if (i >= BB * 2 * HWC) return;
  int b = i / (2 * HWC);
  int r = i - b * 2 * HWC;
  int c = r / HWC;
  int pos = r - c * HWC;
  int y = pos / WW, x = pos - (pos / WW) * WW;
  const float* A = intr + b * 9;
  float fu = A[0] * 0.125f;              // W/ORIG_W
  float cu = (A[2] + 0.5f) * 0.125f;
  float fv = A[4] * 0.125f;              // H/ORIG_H
  float cv = (A[5] + 0.5f) * 0.125f;
  float v = (c == 0) ? (((float)x + 0.5f) - cu) / fu
                     : (((float)y + 0.5f) - cv) / fv;
  uv8[i] = v;
}

// hx130 = concat(h(64), feat(64), uv8(2))
__global__ void k_build130(const float* __restrict__ h, const float* __restrict__ feat,
                           const float* __restrict__ uv8, float* __restrict__ hx) {
  int i = blockIdx.x * blockDim.x + threadIdx.x;
  if (i >= BB * 130 * HWC) return;
  int b = i / (130 * HWC);
  int r = i - b * 130 * HWC;
  int c = r / HWC;
  int pos = r - c * HWC;
  float v;
  if (c < 64)       v = h[(b * 64 + c) * HWC + pos];
  else if (c < 128) v = feat[(b * 64 + (c - 64)) * HWC + pos];
  else              v = uv8[(b * 2 + (c - 128)) * HWC + pos];
  hx[i] = v;
}

// hx130 = concat(r*h, feat, uv8)
__global__ void k_build130_rh(const float* __restrict__ rg, const float* __restrict__ h,
                              const float* __restrict__ feat, const float* __restrict__ uv8,
                              float* __restrict__ hx) {
  int i = blockIdx.x * blockDim.x + threadIdx.x;
  if (i >= BB * 130 * HWC) return;
  int b = i / (130 * HWC);
  int r = i - b * 130 * HWC;
  int c = r / HWC;
  int pos = r - c * HWC;
  float v;
  if (c < 64) {
    int idx = (b * 64 + c) * HWC + pos;
    v = rg[idx] * h[idx];
  } else if (c < 128) v = feat[(b * 64 + (c - 64)) * HWC + pos];
  else                v = uv8[(b * 2 + (c - 128)) * HWC + pos];
  hx[i] = v;
}

// hx66 = concat(h(64), uv8(2))
__global__ void k_build66(const float* __restrict__ h, const float* __restrict__ uv8,
                          float* __restrict__ hx) {
  int i = blockIdx.x * blockDim.x + threadIdx.x;
  if (i >= BB * 66 * HWC) return;
  int b = i / (66 * HWC);
  int r = i - b * 66 * HWC;
  int c = r / HWC;
  int pos = r - c * HWC;
  float v = (c < 64) ? h[(b * 64 + c) * HWC + pos]
                     : uv8[(b * 2 + (c - 64)) * HWC + pos];
  hx[i] = v;
}

// h_new = (1-z)*h + z*q
__global__ void k_gru_h(const float* __restrict__ z, const float* __restrict__ h,
                        const float* __restrict__ q, float* __restrict__ hn) {
  int i = blockIdx.x * blockDim.x + threadIdx.x;
  if (i >= BB * 64 * HWC) return;
  float zz = z[i];
  hn[i] = (1.0f - zz) * h[i] + zz * q[i];
}

// pred4 = concat(pred_norm(3), kappa(1)); kap_is_raw => apply elu(x)+1
__global__ void k_pred4(const float* __restrict__ pn, const float* __restrict__ kap,
                        int kap_is_raw, float* __restrict__ pred4) {
  int i = blockIdx.x * blockDim.x + threadIdx.x;
  if (i >= BB * 4 * HWC) return;
  int b = i / (4 * HWC);
  int r = i - b * 4 * HWC;
  int c = r / HWC;
  int pos = r - c * HWC;
  float v;
  if (c < 3) v = pn[(b * 3 + c) * HWC + pos];
  else {
    float k = kap[b * HWC + pos];
    v = kap_is_raw ? ((k > 0.0f ? k : (expf(k) - 1.0f)) + 1.0f) : k;
  }
  pred4[i] = v;
}

// Rotation-based refinement: per pixel, 5x5 patch loop.
__global__ void k_refine(const float* __restrict__ pn, const float* __restrict__ prob,
                         const float* __restrict__ xys, const float* __restrict__ ang,
                         const float* __restrict__ intr, float* __restrict__ pn_out) {
  int i = blockIdx.x * blockDim.x + threadIdx.x;
  if (i >= MTOT) return;
  int b = i / HWC;
  int pos = i - b * HWC;
  int y = pos / WW, x = pos - (pos / WW) * WW;
  const float* A = intr + b * 9;
  float fu = A[0] * 0.125f;
  float cu = (A[2] + 0.5f) * 0.125f;
  float fv = A[4] * 0.125f;
  float cv = (A[5] + 0.5f) * 0.125f;

  float accx = 0.f, accy = 0.f, accz = 0.f;
#pragma unroll 1
  for (int p = 0; p < 25; ++p) {
    int py = p / 5, px = p - (p / 5) * 5;
    int yy = min(max(y + py - 2, 0), HH - 1);   // edge padding
    int xx = min(max(x + px - 2, 0), WW - 1);
    int pidx = yy * WW + xx;
    float nx = pn[(b * 3 + 0) * HWC + pidx];
    float ny = pn[(b * 3 + 1) * HWC + pidx];
    float nz = pn[(b * 3 + 2) * HWC + pidx];

    float u = xys[(b * 50 + p) * HWC + pos];
    float v = xys[(b * 50 + 25 + p) * HWC + pos];
    float xn = fmaxf(sqrtf(u * u + v * v), 1e-12f);
    u /= xn; v /= xn;

    float theta0 = ang[(b * 25 + p) * HWC + pos] * 3.14159265358979323846f;
    float pr = prob[(b * 25 + p) * HWC + pos];

    float du = u / fu, dv = v / fv;
    float tu = (((float)xx + 0.5f) + u - cu) / fu;
    float tv = (((float)yy + 0.5f) + v - cv) / fv;

    float den = tu * nx + tv * ny + nz;
    if (fabsf(den) < 1e-8f) {
      float sg = (den > 0.f) ? 1.f : ((den < 0.f) ? -1.f : 0.f);
      den = 1e-8f * sg;
    }
    float dz = -(du * nx + dv * ny) / den;
    float ax = du + dz * tu, ay = dv + dz * tv, az = dz;
    float an = sqrtf(ax * ax + ay * ay + az * az);
    float aninv = 1.0f / fmaxf(an, 1e-12f);
    ax *= aninv; ay *= aninv; az *= aninv;
    if (!(__builtin_isfinite(ax) && __builtin_isfinite(ay) && __builtin_isfinite(az))) {
      ax = 0.f; ay = 0.f; az = 0.f;
    }
    float amag = sqrtf(ax * ax + ay * ay + az * az);
    float th = theta0 * amag;
    float s = sinf(th), c = cosf(th);
    float kdv = ax * nx + ay * ny + az * nz;
    float cx = ay * nz - az * ny;
    float cy = az * nx - ax * nz;
    float cz = ax * ny - ay * nx;
    float omc = 1.0f - c;
    float rx = nx * c + cx * s + ax * kdv * omc;
    float ry = ny * c + cy * s + ay * kdv * omc;
    float rz = nz * c + cz * s + az * kdv * omc;
    float rn = 1.0f / fmaxf(sqrtf(rx * rx + ry * ry + rz * rz), 1e-12f);
    accx += pr * rx * rn;
    accy += pr * ry * rn;
    accz += pr * rz * rn;
  }
  float inv = 1.0f / fmaxf(sqrtf(accx * accx + accy * accy + accz * accz), 1e-12f);
  pn_out[(b * 3 + 0) * HWC + pos] = accx * inv;
  pn_out[(b * 3 + 1) * HWC + pos] = accy * inv;
  pn_out[(b * 3 + 2) * HWC + pos] = accz * inv;
}

// Learned-mask 8x upsample + normal_act; writes one pred (B,4,480,640).
__global__ void k_upsample(const float* __restrict__ pred4, const float* __restrict__ up_mask,
                           float* __restrict__ out) {
  int i = blockIdx.x * blockDim.x + threadIdx.x;
  if (i >= BB * UPH * UPW) return;
  int b = i / (UPH * UPW);
  int r = i - b * UPH * UPW;
  int hh = r / UPW;
  int ww = r - hh * UPW;
  int h0 = hh >> 3, ki = hh & 7;
  int w0 = ww >> 3, kj = ww & 7;

  float mv[9], mmax = -3.402823466e38f;
#pragma unroll
  for (int m = 0; m < 9; ++m) {
    int ch = m * 64 + ki * 8 + kj;   // reshape (9,K,K)
    float v = up_mask[((b * 576 + ch) * HH + h0) * WW + w0];
    mv[m] = v;
    mmax = fmaxf(mmax, v);
  }
  float wsum = 0.f;
#pragma unroll
  for (int m = 0; m < 9; ++m) { mv[m] = expf(mv[m] - mmax); wsum += mv[m]; }
  float inv = 1.0f / wsum;

  float o0 = 0.f, o1 = 0.f, o2 = 0.f, o3 = 0.f;
#pragma unroll
  for (int m = 0; m < 9; ++m) {
    int ii = m / 3, jj = m - (m / 3) * 3;
    int yy = min(max(h0 + ii - 1, 0), HH - 1);   // edge-padded 3x3 patch
    int xx = min(max(w0 + jj - 1, 0), WW - 1);
    float wgt = mv[m] * inv;
    o0 += wgt * pred4[((b * 4 + 0) * HH + yy) * WW + xx];
    o1 += wgt * pred4[((b * 4 + 1) * HH + yy) * WW + xx];
    o2 += wgt * pred4[((b * 4 + 2) * HH + yy) * WW + xx];
    o3 += wgt * pred4[((b * 4 + 3) * HH + yy) * WW + xx];
  }
  float nrm = 1.0f / fmaxf(sqrtf(o0 * o0 + o1 * o1 + o2 * o2), 1e-12f);
  out[((b * 4 + 0) * UPH + hh) * UPW + ww] = o0 * nrm;
  out[((b * 4 + 1) * UPH + hh) * UPW + ww] = o1 * nrm;
  out[((b * 4 + 2) * UPH + hh) * UPW + ww] = o2 * nrm;
  out[((b * 4 + 3) * UPH + hh) * UPW + ww] = o3 * nrm;
}

// ---------------------------------------------------------------------------
// Host orchestration
// ---------------------------------------------------------------------------
static inline void head_apply_launch(const float* hx66, float* t1, float* t2, float* outb,
                                     const float* w1, const float* b1,
                                     const float* w2, const float* b2,
                                     const float* w3, const float* b3,
                                     int cout3, int act3, hipStream_t s) {
  conv_wmma<3, 1><<<dim3(MT, 1), 128, 0, s>>>(hx66, w1, b1, t1, 66, 64, ACT_RELU);
  conv_wmma<1, 0><<<dim3(MT, 1), 128, 0, s>>>(t1, w2, b2, t2, 64, 64, ACT_RELU);
  conv_wmma<1, 0><<<dim3(MT, (cout3 + 63) / 64), 128, 0, s>>>(t2, w3, b3, outb, 64, cout3, act3);
}

#define GRID1D(n) dim3((((n) + 255) / 256)), dim3(256)

extern "C" void kernel_launch(void* const* d_in, const int* in_sizes, int n_in,
                              void* d_out, int out_size, void* d_ws, size_t ws_size,
                              hipStream_t stream) {
  (void)in_sizes; (void)n_in; (void)out_size; (void)ws_size;

  // Inputs in setup_inputs() dict order (nested dicts in insertion order).
  const float* pred_norm  = (const float*)d_in[0];
  const float* feat_map   = (const float*)d_in[1];
  const float* h0         = (const float*)d_in[2];
  const float* pred_kappa = (const float*)d_in[3];
  const float* intr       = (const float*)d_in[4];
  const float* wz = (const float*)d_in[5];  const float* bz = (const float*)d_in[6];
  const float* wr = (const float*)d_in[7];  const float* br = (const float*)d_in[8];
  const float* wq = (const float*)d_in[9];  const float* bq = (const float*)d_in[10];
  // heads: w1,b1,w2,b2,w3,b3 starting at 11: prob, xy, angle, kappa, up
  const float* HD[5][6];
  for (int h = 0; h < 5; ++h)
    for (int j = 0; j < 6; ++j)
      HD[h][j] = (const float*)d_in[11 + h * 6 + j];
  const float* const* prob_h  = HD[0];
  const float* const* xy_h    = HD[1];
  const float* const* angle_h = HD[2];
  const float* const* kappa_h = HD[3];
  const float* const* up_h    = HD[4];

  float* ws = (float*)d_ws;
  size_t off = 0;
  auto alloc = [&](size_t n) { float* p = ws + off; off += n; return p; };
  const size_t HN = (size_t)BB * 64 * HWC;
  float* uv8     = alloc((size_t)BB * 2 * HWC);
  float* hx130   = alloc((size_t)BB * 130 * HWC);   // reused for [r*h, feat, uv8]
  float* hx66    = alloc((size_t)BB * 66 * HWC);
  float* zb      = alloc(HN);
  float* rb      = alloc(HN);
  float* qb      = alloc(HN);
  float* hA      = alloc(HN);
  float* hB      = alloc(HN);
  float* t1      = alloc(HN);
  float* t2      = alloc(HN);
  float* up_mask = alloc((size_t)BB * 576 * HWC);
  float* probb   = alloc((size_t)BB * 25 * HWC);
  float* xysb    = alloc((size_t)BB * 50 * HWC);
  float* angb    = alloc((size_t)BB * 25 * HWC);
  float* kapb    = alloc((size_t)BB * 1 * HWC);
  float* pnA     = alloc((size_t)BB * 3 * HWC);
  float* pnB     = alloc((size_t)BB * 3 * HWC);
  float* pred4   = alloc((size_t)BB * 4 * HWC);

  float* outp = (float*)d_out;

  // --- camera grid ---
  k_uv8<<<GRID1D(BB * 2 * HWC), 0, stream>>>(intr, uv8);

  // --- preds[0]: up_head on concat(h0, uv8), upsample concat(pred_norm, pred_kappa)
  k_build66<<<GRID1D(BB * 66 * HWC), 0, stream>>>(h0, uv8, hx66);
  head_apply_launch(hx66, t1, t2, up_mask, up_h[0], up_h[1], up_h[2], up_h[3],
                    up_h[4], up_h[5], 576, ACT_NONE, stream);
  k_pred4<<<GRID1D(BB * 4 * HWC), 0, stream>>>(pred_norm, pred_kappa, 0, pred4);
  k_upsample<<<GRID1D(BB * UPH * UPW), 0, stream>>>(pred4, up_mask, outp);

  // --- state init ---
  hipMemcpyAsync(hA, h0, HN * sizeof(float), hipMemcpyDeviceToDevice, stream);
  hipMemcpyAsync(pnA, pred_norm, (size_t)BB * 3 * HWC * sizeof(float),
                 hipMemcpyDeviceToDevice, stream);
  float* hc = hA; float* hn = hB;
  float* pc = pnA; float* px = pnB;

  for (int it = 0; it < 2; ++it) {
    // GRU
    k_build130<<<GRID1D(BB * 130 * HWC), 0, stream>>>(hc, feat_map, uv8, hx130);
    conv_wmma<5, 2><<<dim3(MT, 1), 128, 0, stream>>>(hx130, wz, bz, zb, 130, 64, ACT_SIG);
    conv_wmma<5, 2><<<dim3(MT, 1), 128, 0, stream>>>(hx130, wr, br, rb, 130, 64, ACT_SIG);
    k_build130_rh<<<GRID1D(BB * 130 * HWC), 0, stream>>>(rb, hc, feat_map, uv8, hx130);
    conv_wmma<5, 2><<<dim3(MT, 1), 128, 0, stream>>>(hx130, wq, bq, qb, 130, 64, ACT_TANH);
    k_gru_h<<<GRID1D(BB * 64 * HWC), 0, stream>>>(zb, hc, qb, hn);

    // heads on concat(h_new, uv8)
    k_build66<<<GRID1D(BB * 66 * HWC), 0, stream>>>(hn, uv8, hx66);
    head_apply_launch(hx66, t1, t2, probb, prob_h[0], prob_h[1], prob_h[2], prob_h[3],
                      prob_h[4], prob_h[5], 25, ACT_SIG, stream);
    head_apply_launch(hx66, t1, t2, kapb, kappa_h[0], kappa_h[1], kappa_h[2], kappa_h[3],
                      kappa_h[4], kappa_h[5], 1, ACT_NONE, stream);
    head_apply_launch(hx66, t1, t2, xysb, xy_h[0], xy_h[1], xy_h[2], xy_h[3],
                      xy_h[4], xy_h[5], 50, ACT_NONE, stream);
    head_apply_launch(hx66, t1, t2, angb, angle_h[0], angle_h[1], angle_h[2], angle_h[3],
                      angle_h[4], angle_h[5], 25, ACT_SIG, stream);

    // rotation refinement
    k_refine<<<GRID1D(MTOT), 0, stream>>>(pc, probb, xysb, angb, intr, px);
    k_pred4<<<GRID1D(BB * 4 * HWC), 0, stream>>>(px, kapb, 1, pred4);

    // upsample with fresh up_mask
    head_apply_launch(hx66, t1, t2, up_mask, up_h[0], up_h[1], up_h[2], up_h[3],
                      up_h[4], up_h[5], 576, ACT_NONE, stream);
    k_upsample<<<GRID1D(BB * UPH * UPW), 0, stream>>>(pred4, up_mask,
                                                     outp + (size_t)(it + 1) * PRED_STRIDE);

    { float* tmp = hc; hc = hn; hn = tmp; }
    { float* tmp = pc; pc = px; px = tmp; }
  }
}